// SpatialGlueMini_68624987455871
// MI455X (gfx1250) — compile-verified
//
#include <hip/hip_runtime.h>
#include <math.h>

typedef __attribute__((ext_vector_type(2))) float v2f;
typedef __attribute__((ext_vector_type(8))) float v8f;

#define NN   50000      // nodes
#define DIN  128
#define DHID 128
#define DLAT 64
#define NE   800000     // explicit edges per graph

// ---------------------------------------------------------------- utilities
__global__ void sg_fill(float* __restrict__ p, float v, int n) {
    int i = blockIdx.x * blockDim.x + threadIdx.x;
    if (i < n) p[i] = v;
}

// deg[col[e]] += 1  (deg buffer pre-filled with 1.0 for the self loop)
__global__ void sg_deg(const int* __restrict__ ei, float* __restrict__ deg, int e) {
    int i = blockIdx.x * blockDim.x + threadIdx.x;
    if (i < e) atomicAdd(&deg[ei[NE + i]], 1.0f);
}

__global__ void sg_rsqrt(float* __restrict__ d, int n) {
    int i = blockIdx.x * blockDim.x + threadIdx.x;
    if (i < n) d[i] = rsqrtf(d[i]);     // deg >= 1 always (self loop)
}

// norm[e] = dis[row[e]] * dis[col[e]]
__global__ void sg_norm(const int* __restrict__ ei, const float* __restrict__ dis,
                        float* __restrict__ nrm, int e) {
    int i = blockIdx.x * blockDim.x + threadIdx.x;
    if (i < e) nrm[i] = dis[ei[i]] * dis[ei[NE + i]];
}

// ------------------------------------------------- fp32 WMMA GEMM  Y = act(X@W + b)
// One wave computes a 16-row stripe x P columns using V_WMMA_F32_16X16X4_F32.
// ACT_IN : 0 = none, 1 = relu applied to A elements on load (fuses relu(prev layer))
// ACT_OUT: 0 = none, 1 = relu, 2 = tanh
template <int K, int P, int ACT_IN, int ACT_OUT, bool HAS_BIAS>
__global__ __launch_bounds__(256) void sg_gemm(const float* __restrict__ X,
                                               const float* __restrict__ W,
                                               const float* __restrict__ bias,
                                               float* __restrict__ Y, int nrows) {
    const int wave = (blockIdx.x * blockDim.x + threadIdx.x) >> 5;
    const int lane = threadIdx.x & 31;
    const int m0   = wave * 16;
    if (m0 >= nrows) return;                 // wave-uniform: EXEC stays all-1s
    const int half = lane >> 4;              // 0: K pair {0,1}; 1: K pair {2,3}
    const int l15  = lane & 15;

    constexpr int NT = P / 16;
    v8f acc[NT] = {};

    const float* __restrict__ xrow = X + (size_t)(m0 + l15) * K;
    for (int k0 = 0; k0 < K; k0 += 4) {
        float ax = xrow[k0 + 2 * half + 0];
        float ay = xrow[k0 + 2 * half + 1];
        if (ACT_IN == 1) { ax = fmaxf(ax, 0.0f); ay = fmaxf(ay, 0.0f); }
        v2f a; a.x = ax; a.y = ay;
#pragma unroll
        for (int t = 0; t < NT; ++t) {
            v2f b;
            b.x = W[(size_t)(k0 + 2 * half + 0) * P + t * 16 + l15];
            b.y = W[(size_t)(k0 + 2 * half + 1) * P + t * 16 + l15];
            acc[t] = __builtin_amdgcn_wmma_f32_16x16x4_f32(
                false, a, false, b, (short)0, acc[t], false, false);
        }
    }
#pragma unroll
    for (int t = 0; t < NT; ++t) {
        const int c  = t * 16 + l15;
        const float bv = HAS_BIAS ? bias[c] : 0.0f;
#pragma unroll
        for (int r = 0; r < 8; ++r) {
            const int row = m0 + r + 8 * half;   // C/D layout: VGPR r -> M=r / M=r+8
            float v = acc[t][r] + bv;
            if (ACT_OUT == 1) v = fmaxf(v, 0.0f);
            else if (ACT_OUT == 2) v = tanhf(v);
            Y[(size_t)row * P + c] = v;
        }
    }
}

// ----------------------------------------- GCN aggregation (segment_sum + self loop)
// init: out[i][c] = xw[i][c]*dis[i]^2 + bias[c]   (self-loop norm = dis^2; bias fused)
template <int P>
__global__ void sg_prop_init(const float* __restrict__ xw, const float* __restrict__ dis,
                             const float* __restrict__ bias, float* __restrict__ out, int n) {
    long i = (long)blockIdx.x * blockDim.x + threadIdx.x;
    if (i >= (long)n * P) return;
    int node = (int)(i / P), c = (int)(i % P);
    float d = dis[node];
    out[i] = xw[i] * d * d + bias[c];
}

// edges: out[col[e]][:] += xw[row[e]][:] * norm[e] ; P/4 lanes per edge, float4 each
template <int P>
__global__ void sg_prop_edges(const float* __restrict__ xw, const int* __restrict__ ei,
                              const float* __restrict__ nrm, float* __restrict__ out, int e) {
    constexpr int TPE = P / 4;
    long gid = (long)blockIdx.x * blockDim.x + threadIdx.x;
    int eidx = (int)(gid / TPE);
    if (eidx >= e) return;
    int c  = (int)(gid % TPE) * 4;
    int r  = ei[eidx];
    int cn = ei[NE + eidx];
    float nv = nrm[eidx];
    float4 v = *(const float4*)(xw + (size_t)r * P + c);
    float* o = out + (size_t)cn * P + c;
    atomicAdd(o + 0, v.x * nv);
    atomicAdd(o + 1, v.y * nv);
    atomicAdd(o + 2, v.z * nv);
    atomicAdd(o + 3, v.w * nv);
}

// ------------------------------------------------- attention helpers
// s[i] = sum_c t[i][c] * q[c], c in [0,64); one wave per node
__global__ void sg_rowdot64(const float* __restrict__ t, const float* __restrict__ q,
                            float* __restrict__ s, int n) {
    int gid  = blockIdx.x * blockDim.x + threadIdx.x;
    int node = gid >> 5, lane = gid & 31;
    if (node >= n) return;
    const float* row = t + (size_t)node * 64;
    float v = row[lane] * q[lane] + row[32 + lane] * q[32 + lane];
#pragma unroll
    for (int off = 16; off >= 1; off >>= 1) v += __shfl_xor(v, off, 32);
    if (lane == 0) s[node] = v;
}

// alpha[i][0..1] = softmax(s0[i], s1[i])
__global__ void sg_softmax2(const float* __restrict__ s0, const float* __restrict__ s1,
                            float* __restrict__ alpha, int n) {
    int i = blockIdx.x * blockDim.x + threadIdx.x;
    if (i >= n) return;
    float a = s0[i], b = s1[i];
    float mx = fmaxf(a, b);
    float e0 = __expf(a - mx), e1 = __expf(b - mx);
    float inv = 1.0f / (e0 + e1);
    alpha[2 * i]     = e0 * inv;
    alpha[2 * i + 1] = e1 * inv;
}

// y[i][c] = alpha[i][0]*h0[i][c] + alpha[i][1]*h1[i][c], c in [0,64)
__global__ void sg_combine64(const float* __restrict__ alpha, const float* __restrict__ h0,
                             const float* __restrict__ h1, float* __restrict__ y, int n) {
    long i = (long)blockIdx.x * blockDim.x + threadIdx.x;
    if (i >= (long)n * 64) return;
    int node = (int)(i >> 6);
    y[i] = alpha[2 * node] * h0[i] + alpha[2 * node + 1] * h1[i];
}

// ---------------------------------------------------------------- host orchestration
static inline int blks(long n, int b) { return (int)((n + b - 1) / b); }

extern "C" void kernel_launch(void* const* d_in, const int* in_sizes, int n_in,
                              void* d_out, int out_size, void* d_ws, size_t ws_size,
                              hipStream_t stream) {
    (void)in_sizes; (void)n_in; (void)out_size; (void)ws_size;
    const int GEMM_GRID = blks(NN / 16, 8);   // 8 waves / 256-thread block

    // ---- inputs (setup_inputs dict order, params flattened in insertion order)
    const float* x[2]     = {(const float*)d_in[0], (const float*)d_in[1]};
    const int*   ei_sp    = (const int*)d_in[2];
    const int*   ei_ft[2] = {(const int*)d_in[3], (const int*)d_in[4]};
    int pi = 5;
    const float *spW1[2], *spB1[2], *spW2[2], *spB2[2];
    const float *ftW1[2], *ftB1[2], *ftW2[2], *ftB2[2];
    for (int m = 0; m < 2; ++m) { spW1[m]=(const float*)d_in[pi++]; spB1[m]=(const float*)d_in[pi++];
                                  spW2[m]=(const float*)d_in[pi++]; spB2[m]=(const float*)d_in[pi++]; }
    for (int m = 0; m < 2; ++m) { ftW1[m]=(const float*)d_in[pi++]; ftB1[m]=(const float*)d_in[pi++];
                                  ftW2[m]=(const float*)d_in[pi++]; ftB2[m]=(const float*)d_in[pi++]; }
    const float *iaW[2], *iaB[2], *iaQ[2];
    for (int m = 0; m < 2; ++m) { iaW[m]=(const float*)d_in[pi++]; iaB[m]=(const float*)d_in[pi++];
                                  iaQ[m]=(const float*)d_in[pi++]; }
    const float *inW=(const float*)d_in[pi++], *inB=(const float*)d_in[pi++], *inQ=(const float*)d_in[pi++];
    const float *dW1[2], *dB1[2], *dW2[2], *dB2[2];
    for (int m = 0; m < 2; ++m) { dW1[m]=(const float*)d_in[pi++]; dB1[m]=(const float*)d_in[pi++];
                                  dW2[m]=(const float*)d_in[pi++]; dB2[m]=(const float*)d_in[pi++]; }

    // ---- outputs: z, ys0, ys1, recon0, recon1, alpha0, alpha1, beta
    float* out = (float*)d_out;
    float* zo      = out;
    float* ys[2]   = {out + (size_t)NN*DLAT, out + 2*(size_t)NN*DLAT};
    float* rec[2]  = {out + 3*(size_t)NN*DLAT, out + 3*(size_t)NN*DLAT + (size_t)NN*DIN};
    float* al[2]   = {out + 3*(size_t)NN*DLAT + 2*(size_t)NN*DIN,
                      out + 3*(size_t)NN*DLAT + 2*(size_t)NN*DIN + 2*(size_t)NN};
    float* beta    =  out + 3*(size_t)NN*DLAT + 2*(size_t)NN*DIN + 4*(size_t)NN;

    // ---- workspace layout
    float* p = (float*)d_ws;
    auto take = [&](size_t cnt) { float* r = p; p += cnt; return r; };
    float* dis_sp = take(NN);  float* dis_ft[2] = {take(NN), take(NN)};
    float* nrm_sp = take(NE);  float* nrm_ft[2] = {take(NE), take(NE)};
    float* bufXW  = take((size_t)NN * DHID);   // x@W1  (also decoder hidden)
    float* bufAG  = take((size_t)NN * DHID);   // layer-1 aggregation
    float* bufXW2 = take((size_t)NN * DLAT);   // relu(a1)@W2
    float* hsp    = take((size_t)NN * DLAT);
    float* hft    = take((size_t)NN * DLAT);
    float* tbuf   = take((size_t)NN * DLAT);   // tanh(h@W+b)
    float* s0     = take(NN);
    float* s1     = take(NN);

    // ---- graph normalization (3 graphs)
    const int* eis[3]  = {ei_sp, ei_ft[0], ei_ft[1]};
    float*     diss[3] = {dis_sp, dis_ft[0], dis_ft[1]};
    float*     nrms[3] = {nrm_sp, nrm_ft[0], nrm_ft[1]};
    for (int g = 0; g < 3; ++g) {
        sg_fill <<<blks(NN, 256), 256, 0, stream>>>(diss[g], 1.0f, NN);  // self loop
        sg_deg  <<<blks(NE, 256), 256, 0, stream>>>(eis[g], diss[g], NE);
        sg_rsqrt<<<blks(NN, 256), 256, 0, stream>>>(diss[g], NN);
        sg_norm <<<blks(NE, 256), 256, 0, stream>>>(eis[g], diss[g], nrms[g], NE);
    }

    // ---- one GCN encoder: h = GCN2( relu(GCN1(x)) )
    auto encoder = [&](const float* xin, const float* W1, const float* b1,
                       const float* W2, const float* b2,
                       const int* ei, const float* dis, const float* nrm, float* h) {
        sg_gemm<DIN, DHID, 0, 0, false><<<GEMM_GRID, 256, 0, stream>>>(xin, W1, nullptr, bufXW, NN);
        sg_prop_init<DHID><<<blks((long)NN * DHID, 256), 256, 0, stream>>>(bufXW, dis, b1, bufAG, NN);
        sg_prop_edges<DHID><<<blks((long)NE * (DHID / 4), 256), 256, 0, stream>>>(bufXW, ei, nrm, bufAG, NE);
        sg_gemm<DHID, DLAT, 1, 0, false><<<GEMM_GRID, 256, 0, stream>>>(bufAG, W2, nullptr, bufXW2, NN);
        sg_prop_init<DLAT><<<blks((long)NN * DLAT, 256), 256, 0, stream>>>(bufXW2, dis, b2, h, NN);
        sg_prop_edges<DLAT><<<blks((long)NE * (DLAT / 4), 256), 256, 0, stream>>>(bufXW2, ei, nrm, h, NE);
    };

    // attention score: s = tanh(h@W + b) @ q
    auto attn = [&](const float* h, const float* W, const float* b, const float* q, float* s) {
        sg_gemm<DLAT, DLAT, 0, 2, true><<<GEMM_GRID, 256, 0, stream>>>(h, W, b, tbuf, NN);
        sg_rowdot64<<<blks((long)NN * 32, 256), 256, 0, stream>>>(tbuf, q, s, NN);
    };

    // ---- per-modality: encoders + intra attention
    for (int m = 0; m < 2; ++m) {
        encoder(x[m], spW1[m], spB1[m], spW2[m], spB2[m], ei_sp, dis_sp, nrm_sp, hsp);
        encoder(x[m], ftW1[m], ftB1[m], ftW2[m], ftB2[m], ei_ft[m], dis_ft[m], nrm_ft[m], hft);
        attn(hsp, iaW[m], iaB[m], iaQ[m], s0);
        attn(hft, iaW[m], iaB[m], iaQ[m], s1);
        sg_softmax2 <<<blks(NN, 256), 256, 0, stream>>>(s0, s1, al[m], NN);
        sg_combine64<<<blks((long)NN * DLAT, 256), 256, 0, stream>>>(al[m], hsp, hft, ys[m], NN);
    }

    // ---- inter attention -> z
    attn(ys[0], inW, inB, inQ, s0);
    attn(ys[1], inW, inB, inQ, s1);
    sg_softmax2 <<<blks(NN, 256), 256, 0, stream>>>(s0, s1, beta, NN);
    sg_combine64<<<blks((long)NN * DLAT, 256), 256, 0, stream>>>(beta, ys[0], ys[1], zo, NN);

    // ---- decoders: recon = relu(z@W1+b1)@W2 + b2
    for (int m = 0; m < 2; ++m) {
        sg_gemm<DLAT, DHID, 0, 1, true><<<GEMM_GRID, 256, 0, stream>>>(zo, dW1[m], dB1[m], bufXW, NN);
        sg_gemm<DHID, DIN, 0, 0, true><<<GEMM_GRID, 256, 0, stream>>>(bufXW, dW2[m], dB2[m], rec[m], NN);
    }
}